// Quanv2d_72584947302551
// MI455X (gfx1250) — compile-verified
//
#include <hip/hip_runtime.h>

typedef __attribute__((ext_vector_type(2))) float v2f;
typedef __attribute__((ext_vector_type(8))) float v8f;

#define LDS_STRIDE 17   // pad 16 -> 17 to avoid LDS bank conflicts

__device__ __forceinline__ void apply_rz(float th, float& r0, float& i0, float& r1, float& i1) {
    float s, c; __sincosf(0.5f * th, &s, &c);
    float nr0 = r0 * c + i0 * s, ni0 = i0 * c - r0 * s;   // * (c - i s)
    float nr1 = r1 * c - i1 * s, ni1 = i1 * c + r1 * s;   // * (c + i s)
    r0 = nr0; i0 = ni0; r1 = nr1; i1 = ni1;
}

__device__ __forceinline__ void apply_ry(float th, float& r0, float& i0, float& r1, float& i1) {
    float s, c; __sincosf(0.5f * th, &s, &c);
    float nr0 = c * r0 - s * r1, ni0 = c * i0 - s * i1;
    float nr1 = s * r0 + c * r1, ni1 = s * i0 + c * i1;
    r0 = nr0; i0 = ni0; r1 = nr1; i1 = ni1;
}

__global__ __launch_bounds__(256) void quanv2d_kernel(
    const float* __restrict__ X, const float* __restrict__ W,
    float* __restrict__ out, int P)
{
    // per-block LDS: 256 patches x 17 floats x {re,im}
    __shared__ float s_re[256 * LDS_STRIDE];
    __shared__ float s_im[256 * LDS_STRIDE];

    const int tid  = threadIdx.x;
    const int lane = tid & 31;
    const int wv   = tid >> 5;
    const int n    = lane & 15;          // WMMA N / M index within half-wave
    const int kh   = (lane >> 4) << 1;   // K sub-offset: 0 for lanes 0-15, 2 for lanes 16-31

    int p = blockIdx.x * 256 + tid;
    const bool active = (p < P);
    const int pe = active ? p : (P - 1);

    // decode patch -> (batch, oi, oj); patch pixel origin = (2*oi, 2*oj)
    int b   = pe / 36;
    int r36 = pe - b * 36;
    int oi  = r36 / 6;
    int oj  = r36 - oi * 6;
    const float* xp = X + (size_t)b * 196 + (oi * 2) * 14 + (oj * 2);

    float ang[16];
    #pragma unroll
    for (int r = 0; r < 4; ++r)
        #pragma unroll
        for (int s = 0; s < 4; ++s)
            ang[r * 4 + s] = xp[r * 14 + s];

    float w[8];
    #pragma unroll
    for (int q = 0; q < 8; ++q) w[q] = W[q];

    // ---- single-qubit states: v[q] = chain applied to |+> ----
    const float H = 0.70710678118654752f;
    float vr[4][2], vi[4][2];
    #pragma unroll
    for (int q = 0; q < 3; ++q) {
        float r0 = H, i0 = 0.f, r1 = H, i1 = 0.f;
        apply_rz(ang[q * 5 + 0], r0, i0, r1, i1);
        apply_ry(ang[q * 5 + 1], r0, i0, r1, i1);
        apply_rz(ang[q * 5 + 2], r0, i0, r1, i1);
        apply_ry(ang[q * 5 + 3], r0, i0, r1, i1);
        apply_rz(ang[q * 5 + 4], r0, i0, r1, i1);
        vr[q][0] = r0; vi[q][0] = i0; vr[q][1] = r1; vi[q][1] = i1;
    }
    {   // qubit 3: Rz(ang[15]) |+>
        float s, c; __sincosf(0.5f * ang[15], &s, &c);
        vr[3][0] = H * c; vi[3][0] = -H * s;
        vr[3][1] = H * c; vi[3][1] =  H * s;
    }

    // ---- diagonal edge phases: e^{+0.5 i * expnt[s]} (patch-independent values) ----
    float ph_r[16], ph_i[16];
    #pragma unroll
    for (int s_ = 0; s_ < 16; ++s_) {
        float b0 = (float)((s_ >> 3) & 1), b1 = (float)((s_ >> 2) & 1);
        float b2 = (float)((s_ >> 1) & 1), b3 = (float)(s_ & 1);
        float e = w[0] * b0 * (2.f * b1 - 1.f)
                + w[1] * b1 * (2.f * b2 - 1.f)
                + w[2] * b2 * (2.f * b3 - 1.f)
                + w[3] * b3 * (2.f * b0 - 1.f);
        __sincosf(0.5f * e, &ph_i[s_], &ph_r[s_]);
    }

    // ---- product state * phases -> LDS (row = tid, col = basis state) ----
    {
        float ur[4], ui[4];
        #pragma unroll
        for (int cd = 0; cd < 4; ++cd) {
            int c = cd >> 1, d = cd & 1;
            ur[cd] = vr[2][c] * vr[3][d] - vi[2][c] * vi[3][d];
            ui[cd] = vr[2][c] * vi[3][d] + vi[2][c] * vr[3][d];
        }
        #pragma unroll
        for (int ab = 0; ab < 4; ++ab) {
            int a = ab >> 1, bq = ab & 1;
            float tr = vr[0][a] * vr[1][bq] - vi[0][a] * vi[1][bq];
            float ti = vr[0][a] * vi[1][bq] + vi[0][a] * vr[1][bq];
            #pragma unroll
            for (int cd = 0; cd < 4; ++cd) {
                int s_ = ab * 4 + cd;
                float ar = tr * ur[cd] - ti * ui[cd];
                float ai = tr * ui[cd] + ti * ur[cd];
                s_re[tid * LDS_STRIDE + s_] = ar * ph_r[s_] - ai * ph_i[s_];
                s_im[tid * LDS_STRIDE + s_] = ar * ph_i[s_] + ai * ph_r[s_];
            }
        }
    }

    // ---- B fragments: B[k][n] = M16[n][k] = prod_q Rq[n_q][k_q], Rq = Ry(w[4+q]) ----
    float cw[4], sw[4];
    #pragma unroll
    for (int q = 0; q < 4; ++q) __sincosf(0.5f * w[4 + q], &sw[q], &cw[q]);
    v2f bf[4];
    #pragma unroll
    for (int c = 0; c < 4; ++c) {
        #pragma unroll
        for (int j = 0; j < 2; ++j) {
            int k = c * 4 + kh + j;
            float val = 1.f;
            #pragma unroll
            for (int q = 0; q < 4; ++q) {
                int nb = (n >> (3 - q)) & 1;
                int kb = (k >> (3 - q)) & 1;
                val *= (nb == kb) ? cw[q] : (nb ? sw[q] : -sw[q]);
            }
            bf[c][j] = val;
        }
    }

    __syncthreads();

    // ---- basis change via V_WMMA_F32_16X16X4_F32: S' = S x M16^T, K accumulated in 4 chunks ----
    v8f dre[2], dim_[2];
    #pragma unroll
    for (int t = 0; t < 2; ++t) {
        v8f acc_r = {0.f, 0.f, 0.f, 0.f, 0.f, 0.f, 0.f, 0.f};
        v8f acc_i = {0.f, 0.f, 0.f, 0.f, 0.f, 0.f, 0.f, 0.f};
        const int rowbase = (wv * 32 + t * 16 + n) * LDS_STRIDE + kh;
        #pragma unroll
        for (int c = 0; c < 4; ++c) {
            v2f a_re, a_im;
            a_re.x = s_re[rowbase + c * 4];
            a_re.y = s_re[rowbase + c * 4 + 1];
            a_im.x = s_im[rowbase + c * 4];
            a_im.y = s_im[rowbase + c * 4 + 1];
            acc_r = __builtin_amdgcn_wmma_f32_16x16x4_f32(false, a_re, false, bf[c],
                                                          (short)0, acc_r, false, false);
            acc_i = __builtin_amdgcn_wmma_f32_16x16x4_f32(false, a_im, false, bf[c],
                                                          (short)0, acc_i, false, false);
        }
        dre[t] = acc_r; dim_[t] = acc_i;
    }

    __syncthreads();

    // ---- probs back to LDS (D-layout: lane holds col n, rows (lane>=16?8:0)+v) ----
    #pragma unroll
    for (int t = 0; t < 2; ++t) {
        const int row0 = wv * 32 + t * 16 + ((lane >> 4) << 3);
        #pragma unroll
        for (int v = 0; v < 8; ++v) {
            float pr = dre[t][v] * dre[t][v] + dim_[t][v] * dim_[t][v];
            s_re[(row0 + v) * LDS_STRIDE + n] = pr;
        }
    }

    __syncthreads();

    // ---- Z-expectation reduction: out[q] = sum_s probs[s] * (1 - 2*bit_q(s)) ----
    float o0 = 0.f, o1 = 0.f, o2 = 0.f, o3 = 0.f;
    #pragma unroll
    for (int s_ = 0; s_ < 16; ++s_) {
        float pr = s_re[tid * LDS_STRIDE + s_];
        o0 += ((s_ >> 3) & 1) ? -pr : pr;
        o1 += ((s_ >> 2) & 1) ? -pr : pr;
        o2 += ((s_ >> 1) & 1) ? -pr : pr;
        o3 += (s_ & 1)        ? -pr : pr;
    }

    if (active) {
        float4 res = make_float4(o0, o1, o2, o3);
        *reinterpret_cast<float4*>(out + (size_t)p * 4) = res;   // out[p*4 + q]
    }
}

extern "C" void kernel_launch(void* const* d_in, const int* in_sizes, int n_in,
                              void* d_out, int out_size, void* d_ws, size_t ws_size,
                              hipStream_t stream) {
    const float* X = (const float*)d_in[0];   // (bs, 1, 14, 14) f32
    const float* W = (const float*)d_in[1];   // (1, 8) f32
    float* out = (float*)d_out;               // (bs, 4, 6, 6) f32 == flat [p*4+q]

    int bs = in_sizes[0] / 196;
    int P  = bs * 36;
    int blocks = (P + 255) / 256;             // bs=16384 -> 2304 full blocks
    quanv2d_kernel<<<blocks, 256, 0, stream>>>(X, W, out, P);
}